// EnhancedSAGE_5257039970570
// MI455X (gfx1250) — compile-verified
//
#include <hip/hip_runtime.h>
#include <hip/hip_bf16.h>
#include <math.h>

// ---------------------------------------------------------------------------
// GraphSAGE 3-layer forward for MI455X (gfx1250, wave32, WMMA)
// N=50000 nodes, E=600000 edges, F=128, H=128, C=47
// ---------------------------------------------------------------------------

#define NN 50000
#define EE 600000
#define FF 128
#define HH 128
#define CC 47
#define CCP 64          // output-layer columns padded to 4 tiles
#define BN_EPS 1e-5f

typedef __attribute__((ext_vector_type(2))) float v2f;
typedef __attribute__((ext_vector_type(8))) float v8f;

// ---------------- generic zero ----------------
__global__ void zero_kernel(float* __restrict__ p, int n) {
    int i = blockIdx.x * blockDim.x + threadIdx.x;
    if (i < n) p[i] = 0.0f;
}

// ---------------- degree count (once, reused by all layers) ----------------
__global__ void count_kernel(const int* __restrict__ dst, float* __restrict__ cnt, int E) {
    int i = blockIdx.x * blockDim.x + threadIdx.x;
    if (i < E) atomicAdd(&cnt[dst[i]], 1.0f);
}

__global__ void inv_kernel(float* __restrict__ cnt, int n) {
    int i = blockIdx.x * blockDim.x + threadIdx.x;
    if (i < n) cnt[i] = 1.0f / fmaxf(cnt[i], 1.0f);
}

// ---------------- weight repack into WMMA B-fragment order ----------------
// pack[((kk*ncolsP + col)*2 + half)*2 + t] = W[(kk*4 + half*2 + t)][col]  (0 if col>=ncols)
// => each lane's B fragment (b.x,b.y) is one contiguous 8-byte load.
__global__ void pack_w_kernel(const float* __restrict__ W, float* __restrict__ pack,
                              int ncols, int ncolsP) {
    int i = blockIdx.x * blockDim.x + threadIdx.x;
    int total = 32 * ncolsP * 4;
    if (i >= total) return;
    int t    = i & 1;
    int half = (i >> 1) & 1;
    int col  = (i >> 2) % ncolsP;
    int kk   = (i >> 2) / ncolsP;
    int k    = kk * 4 + half * 2 + t;
    pack[i] = (col < ncols) ? W[(size_t)k * ncols + col] : 0.0f;
}

__global__ void pack_b_kernel(const float* __restrict__ b, float* __restrict__ bp,
                              int ncols, int ncolsP) {
    int i = blockIdx.x * blockDim.x + threadIdx.x;
    if (i < ncolsP) bp[i] = (i < ncols) ? b[i] : 0.0f;
}

// ---------------- edge scatter: agg[dst] += feat[src] ----------------
// one thread per (edge, 4-float chunk); 32 chunks cover 128 features
__global__ void scatter_kernel(const float* __restrict__ feat,
                               const int* __restrict__ srcIdx,
                               const int* __restrict__ dstIdx,
                               float* __restrict__ agg, int E) {
    int tid = blockIdx.x * blockDim.x + threadIdx.x;
    int e = tid >> 5;
    if (e >= E) return;
    int chunk = (tid & 31) << 2;              // feature offset 0..124
    int s = srcIdx[e];
    int d = dstIdx[e];
    float4 v = *reinterpret_cast<const float4*>(feat + (size_t)s * FF + chunk);
    float* dp = agg + (size_t)d * FF + chunk;
    atomicAdd(dp + 0, v.x);
    atomicAdd(dp + 1, v.y);
    atomicAdd(dp + 2, v.z);
    atomicAdd(dp + 3, v.w);
}

// ---------------- fused SAGE GEMM via WMMA f32 16x16x4 ----------------
// out[M x ncolsOut] = (agg * cnt_inv) @ Wl + bl + xin @ Wr
// One wave computes a 16x64 output slab (4 dense column tiles). Weights are
// pre-packed so every B fragment is one unconditional b64 at a literal offset
// from a running per-lane pointer; all 10 loads of a k-step are hoisted ahead
// of the 8 WMMAs so they clause and the waits stagger. No EXEC divergence in
// the compute loop (stores alone are guarded).
// A layout (16x4 f32): lanes 0-15 -> M=lane, K={0,1}; lanes 16-31 -> M=lane-16, K={2,3}
// B layout (4x16 f32): lanes 0-15 -> N=lane, K={0,1}; lanes 16-31 -> N=lane-16, K={2,3}
// C/D layout: VGPR r: lanes 0-15 -> M=r, lanes 16-31 -> M=8+r
__global__ __launch_bounds__(128)
void sage_gemm_kernel(const float* __restrict__ agg,
                      const float* __restrict__ xin,
                      const float* __restrict__ cnt_inv,
                      const float* __restrict__ pWl,   // packed
                      const float* __restrict__ blP,   // padded bias
                      const float* __restrict__ pWr,   // packed
                      float* __restrict__ out,
                      int ncolsP, int ncolsOut, int ldout, int nGroups) {
    const int K = 128;
    int wave = blockIdx.x * (blockDim.x >> 5) + (threadIdx.x >> 5);
    int lane = threadIdx.x & 31;
    const int ntilesM = NN >> 4;              // 3125, exact
    if (wave >= ntilesM * nGroups) return;
    int tm  = wave / nGroups;
    int tng = wave - tm * nGroups;            // column-tile group (4 dense tiles)

    int half = lane >> 4;                     // 0 or 1
    int l16  = lane & 15;
    int row  = tm * 16 + l16;                 // A row handled by this lane
    int koff = half * 2;                      // this lane's K sub-pair

    float scale = cnt_inv[row];               // mean-aggregation divisor

    int col0 = tng * 64 + l16;                // first of 4 cols (stride 16)
    v8f acc[4];
#pragma unroll
    for (int j = 0; j < 4; ++j) {
        float bias = blP[col0 + 16 * j];
#pragma unroll
        for (int r = 0; r < 8; ++r) acc[j][r] = bias;
    }

    const float* arow = agg + (size_t)row * K + koff;
    const float* xrow = xin + (size_t)row * K + koff;

    // running per-lane packed-B pointers; the 4 tiles sit at literal offsets
    // 0/64/128/192 floats (256-byte stride). Advance by ncolsP*4 per k-step.
    const size_t bstep = (size_t)ncolsP * 4;
    const float* plp = pWl + ((size_t)col0 * 2 + half) * 2;
    const float* prp = pWr + ((size_t)col0 * 2 + half) * 2;

#pragma unroll 2
    for (int kk = 0; kk < K / 4; ++kk) {
        // ---- hoisted loads (clause-friendly) ----
        v2f a1 = *reinterpret_cast<const v2f*>(arow + kk * 4);
        v2f a2 = *reinterpret_cast<const v2f*>(xrow + kk * 4);
        v2f b1[4], b2[4];
#pragma unroll
        for (int j = 0; j < 4; ++j) {
            b1[j] = *reinterpret_cast<const v2f*>(plp + 64 * j);
            b2[j] = *reinterpret_cast<const v2f*>(prp + 64 * j);
        }
        plp += bstep;
        prp += bstep;
        a1.x *= scale;
        a1.y *= scale;

        // ---- 8 WMMAs, 4 independent accumulation chains ----
#pragma unroll
        for (int j = 0; j < 4; ++j) {
            acc[j] = __builtin_amdgcn_wmma_f32_16x16x4_f32(false, a1, false, b1[j],
                                                           (short)0, acc[j], false, false);
        }
#pragma unroll
        for (int j = 0; j < 4; ++j) {
            acc[j] = __builtin_amdgcn_wmma_f32_16x16x4_f32(false, a2, false, b2[j],
                                                           (short)0, acc[j], false, false);
        }
    }

    int baseRow = tm * 16 + half * 8;
#pragma unroll
    for (int j = 0; j < 4; ++j) {
        int col = col0 + 16 * j;
        if (col < ncolsOut) {
#pragma unroll
            for (int r = 0; r < 8; ++r) {
                out[(size_t)(baseRow + r) * ldout + col] = acc[j][r];
            }
        }
    }
}

// ---------------- batchnorm statistics ----------------
// blockDim = 128 (one thread per column), blocks stride over rows (coalesced)
__global__ void bn_stats_kernel(const float* __restrict__ h,
                                float* __restrict__ sum,
                                float* __restrict__ sumsq, int nrows) {
    int col = threadIdx.x;
    float s = 0.0f, s2 = 0.0f;
    for (int r = blockIdx.x; r < nrows; r += gridDim.x) {
        float v = h[(size_t)r * HH + col];
        s += v;
        s2 += v * v;
    }
    atomicAdd(&sum[col], s);
    atomicAdd(&sumsq[col], s2);
}

__global__ void bn_finalize_kernel(const float* __restrict__ sum,
                                   const float* __restrict__ sumsq,
                                   float* __restrict__ mean,
                                   float* __restrict__ inv_std, int nrows) {
    int col = threadIdx.x;
    if (col < HH) {
        float mu = sum[col] / (float)nrows;
        float var = sumsq[col] / (float)nrows - mu * mu;
        mean[col] = mu;
        inv_std[col] = rsqrtf(var + BN_EPS);
    }
}

__global__ void bn_relu_kernel(float* __restrict__ h,
                               const float* __restrict__ mean,
                               const float* __restrict__ inv_std,
                               const float* __restrict__ gamma,
                               const float* __restrict__ beta, int n) {
    int i = blockIdx.x * blockDim.x + threadIdx.x;
    if (i >= n) return;
    int col = i & (HH - 1);
    float v = (h[i] - mean[col]) * inv_std[col] * gamma[col] + beta[col];
    h[i] = fmaxf(v, 0.0f);
}

// ---------------- log_softmax over 47 columns, one wave per row ----------------
__global__ void lsm_kernel(float* __restrict__ out, int nrows, int ncols) {
    int wid = blockIdx.x * (blockDim.x >> 5) + (threadIdx.x >> 5);
    int lane = threadIdx.x & 31;
    if (wid >= nrows) return;
    float* row = out + (size_t)wid * ncols;
    float v0 = (lane < ncols)      ? row[lane]      : -INFINITY;
    float v1 = (lane + 32 < ncols) ? row[lane + 32] : -INFINITY;
    float m = fmaxf(v0, v1);
#pragma unroll
    for (int off = 16; off > 0; off >>= 1) m = fmaxf(m, __shfl_xor(m, off, 32));
    float s = ((lane < ncols) ? expf(v0 - m) : 0.0f) +
              ((lane + 32 < ncols) ? expf(v1 - m) : 0.0f);
#pragma unroll
    for (int off = 16; off > 0; off >>= 1) s += __shfl_xor(s, off, 32);
    float lse = m + logf(s);
    if (lane < ncols) row[lane] = v0 - lse;
    if (lane + 32 < ncols) row[lane + 32] = v1 - lse;
}

// ---------------------------------------------------------------------------
extern "C" void kernel_launch(void* const* d_in, const int* in_sizes, int n_in,
                              void* d_out, int out_size, void* d_ws, size_t ws_size,
                              hipStream_t stream) {
    const float* x   = (const float*)d_in[0];
    const int* edge  = (const int*)d_in[1];
    const float* Wl0 = (const float*)d_in[2];
    const float* bl0 = (const float*)d_in[3];
    const float* Wr0 = (const float*)d_in[4];
    const float* g0  = (const float*)d_in[5];
    const float* b0  = (const float*)d_in[6];
    const float* Wl1 = (const float*)d_in[7];
    const float* bl1 = (const float*)d_in[8];
    const float* Wr1 = (const float*)d_in[9];
    const float* g1  = (const float*)d_in[10];
    const float* b1  = (const float*)d_in[11];
    const float* Wlo = (const float*)d_in[12];
    const float* blo = (const float*)d_in[13];
    const float* Wro = (const float*)d_in[14];

    const int* srcIdx = edge;            // edge_index[0]
    const int* dstIdx = edge + EE;       // edge_index[1]

    // workspace carve-up
    char* ws = (char*)d_ws;
    const size_t featBytes = (size_t)NN * FF * sizeof(float);   // 25.6 MB
    float* bufA    = (float*)(ws);                               // agg
    float* bufB    = (float*)(ws + featBytes);                   // h0
    float* bufC    = (float*)(ws + 2 * featBytes);               // h1
    float* cnt_inv = (float*)(ws + 3 * featBytes);               // N floats
    float* stats   = (float*)(ws + 3 * featBytes + ((size_t)NN * 4 + 255) / 256 * 256);
    float* sum     = stats;              // 128
    float* sumsq   = stats + 128;        // 128
    float* mean    = stats + 256;        // 128
    float* inv_std = stats + 384;        // 128

    const int packHn = 32 * HH * 4;      // 16384 floats per H-layer weight
    const int packCn = 32 * CCP * 4;     // 8192 floats per output weight
    float* pWl0 = stats + 512;
    float* pWr0 = pWl0 + packHn;
    float* pWl1 = pWr0 + packHn;
    float* pWr1 = pWl1 + packHn;
    float* pWlo = pWr1 + packHn;
    float* pWro = pWlo + packCn;
    float* blP0 = pWro + packCn;         // 128
    float* blP1 = blP0 + HH;             // 128
    float* blPo = blP1 + HH;             // 64

    const int T = 256;
    dim3 blk(T);
    int nFeat = NN * FF;                                 // 6.4M
    dim3 gFeat((nFeat + T - 1) / T);
    dim3 gN((NN + T - 1) / T);
    dim3 gE((EE + T - 1) / T);
    dim3 gScat(((size_t)EE * 32 + T - 1) / T);
    dim3 gStats(256);
    dim3 gPackH((packHn + T - 1) / T), gPackC((packCn + T - 1) / T);

    // GEMM grids: 4 waves per 128-thread block, one 16x64 slab per wave
    const int ntilesM = NN / 16;                         // 3125
    const int nGroupsH = HH / 64;                        // 2
    const int nGroupsC = CCP / 64;                       // 1
    int wavesH = ntilesM * nGroupsH;                     // 6250
    int wavesC = ntilesM * nGroupsC;                     // 3125
    dim3 gGemmH((wavesH + 3) / 4), gGemmC((wavesC + 3) / 4), blkGemm(128);

    // -------- weight repack (tiny, once per launch) --------
    pack_w_kernel<<<gPackH, blk, 0, stream>>>(Wl0, pWl0, HH, HH);
    pack_w_kernel<<<gPackH, blk, 0, stream>>>(Wr0, pWr0, HH, HH);
    pack_w_kernel<<<gPackH, blk, 0, stream>>>(Wl1, pWl1, HH, HH);
    pack_w_kernel<<<gPackH, blk, 0, stream>>>(Wr1, pWr1, HH, HH);
    pack_w_kernel<<<gPackC, blk, 0, stream>>>(Wlo, pWlo, CC, CCP);
    pack_w_kernel<<<gPackC, blk, 0, stream>>>(Wro, pWro, CC, CCP);
    pack_b_kernel<<<dim3(1), dim3(128), 0, stream>>>(bl0, blP0, HH, HH);
    pack_b_kernel<<<dim3(1), dim3(128), 0, stream>>>(bl1, blP1, HH, HH);
    pack_b_kernel<<<dim3(1), dim3(64), 0, stream>>>(blo, blPo, CC, CCP);

    // -------- degree (once) --------
    zero_kernel<<<gN, blk, 0, stream>>>(cnt_inv, NN);
    count_kernel<<<gE, blk, 0, stream>>>(dstIdx, cnt_inv, EE);
    inv_kernel<<<gN, blk, 0, stream>>>(cnt_inv, NN);

    // -------- layer 0: x -> bufB --------
    zero_kernel<<<gFeat, blk, 0, stream>>>(bufA, nFeat);
    scatter_kernel<<<gScat, blk, 0, stream>>>(x, srcIdx, dstIdx, bufA, EE);
    sage_gemm_kernel<<<gGemmH, blkGemm, 0, stream>>>(bufA, x, cnt_inv, pWl0, blP0, pWr0,
                                                     bufB, HH, HH, HH, nGroupsH);
    zero_kernel<<<dim3(1), blk, 0, stream>>>(stats, 256);
    bn_stats_kernel<<<gStats, dim3(128), 0, stream>>>(bufB, sum, sumsq, NN);
    bn_finalize_kernel<<<dim3(1), dim3(128), 0, stream>>>(sum, sumsq, mean, inv_std, NN);
    bn_relu_kernel<<<gFeat, blk, 0, stream>>>(bufB, mean, inv_std, g0, b0, nFeat);

    // -------- layer 1: bufB -> bufC --------
    zero_kernel<<<gFeat, blk, 0, stream>>>(bufA, nFeat);
    scatter_kernel<<<gScat, blk, 0, stream>>>(bufB, srcIdx, dstIdx, bufA, EE);
    sage_gemm_kernel<<<gGemmH, blkGemm, 0, stream>>>(bufA, bufB, cnt_inv, pWl1, blP1, pWr1,
                                                     bufC, HH, HH, HH, nGroupsH);
    zero_kernel<<<dim3(1), blk, 0, stream>>>(stats, 256);
    bn_stats_kernel<<<gStats, dim3(128), 0, stream>>>(bufC, sum, sumsq, NN);
    bn_finalize_kernel<<<dim3(1), dim3(128), 0, stream>>>(sum, sumsq, mean, inv_std, NN);
    bn_relu_kernel<<<gFeat, blk, 0, stream>>>(bufC, mean, inv_std, g1, b1, nFeat);

    // -------- output layer: bufC -> d_out (N x 47), then log_softmax --------
    float* out = (float*)d_out;
    zero_kernel<<<gFeat, blk, 0, stream>>>(bufA, nFeat);
    scatter_kernel<<<gScat, blk, 0, stream>>>(bufC, srcIdx, dstIdx, bufA, EE);
    sage_gemm_kernel<<<gGemmC, blkGemm, 0, stream>>>(bufA, bufC, cnt_inv, pWlo, blPo, pWro,
                                                     out, CCP, CC, CC, nGroupsC);
    lsm_kernel<<<dim3((NN + 3) / 4), dim3(128), 0, stream>>>(out, NN, CC);
}